// Psf_layer_22101901705727
// MI455X (gfx1250) — compile-verified
//
#include <hip/hip_runtime.h>

// ---------------------------------------------------------------------------
// PSF layer pipeline for MI455X (gfx1250, wave32).
// FFTs: four-step Cooley-Tukey, both DFT stages as fp32 complex matmuls on
// V_WMMA_F32_16X16X4_F32. One wave = one FFT line (persistent: each wave
// loops over lines, DFT/twiddle fragments hoisted into VGPRs once per wave).
// All intermediates live in d_ws and stay L2-resident (192 MB L2 >> 61 MB).
// ---------------------------------------------------------------------------

typedef __attribute__((ext_vector_type(2))) float v2f;
typedef __attribute__((ext_vector_type(8))) float v8f;

#define PI2F 6.28318530717958647692f

__device__ __forceinline__ v8f wmma4(v2f a, v2f b, v8f c) {
  // D = A(16x4 f32) * B(4x16 f32) + C(16x16 f32)
  return __builtin_amdgcn_wmma_f32_16x16x4_f32(
      false, a, false, b, (short)0, c, false, false);
}

// twiddle W^{prod} for modulus Nmod (power of two), sign sgn (-1 fwd, +1 inv)
__device__ __forceinline__ float2 twf(int prod, int Nmod, float sgn) {
  int m = prod & (Nmod - 1);
  if (m > (Nmod >> 1)) m -= Nmod;
  float a = sgn * (PI2F / (float)Nmod) * (float)m;
  float s, c;
  __sincosf(a, &s, &c);
  return make_float2(c, s);
}

// ---------------------------------------------------------------------------
// In-place batched 1D FFT pass. N = 16*N2 (N2 = 32 -> 512pt, N2 = 16 -> 256pt)
// One wave per line, persistent over multiple lines; 8 waves per 256-thr block.
// line l: base = (l / linesPerImage)*imageStride + (l % linesPerImage)*lineStride
// element i at base + i*elemStride.  Waves are fully independent (private LDS
// slice, per-wave DS ordering) -> no block barriers, only compiler fences.
// ---------------------------------------------------------------------------
template <int N2>
__global__ __launch_bounds__(256) void k_fft_pass(
    float2* buf, int linesPerImage, long long imageStride,
    long long lineStride, long long elemStride, float sgn, float scale,
    int nlines) {
  constexpr int N = 16 * N2;
  constexpr int TILES = N2 / 16;
  constexpr int CH_A = N2 / 4;  // K chunks in step A
  __shared__ float2 shb[8][N];

  const int wave = threadIdx.x >> 5;
  const int lane = threadIdx.x & 31;
  const int lr = lane & 15;
  const int hi2 = (lane >> 4) << 1;  // 0 or 2 (K offset of upper half-wave)
  const int hi8 = (lane >> 4) << 3;  // 0 or 8 (M/row offset of upper half)
  float2* lds = shb[wave];

  // ---- hoisted, loop-invariant fragments (live in VGPRs) ------------------
  // Step A: B = DFT_{N2}: W_{N2}^{sgn*n2*k2}, n2 = K, k2 = N column
  v2f brA[CH_A][TILES], biA[CH_A][TILES];
#pragma unroll
  for (int c = 0; c < CH_A; ++c) {
    const int kk = 4 * c + hi2;
#pragma unroll
    for (int t = 0; t < TILES; ++t) {
      const int n = lr + 16 * t;
      float2 w0 = twf(kk * n, N2, sgn);
      float2 w1 = twf((kk + 1) * n, N2, sgn);
      brA[c][t] = v2f{w0.x, w1.x};
      biA[c][t] = v2f{w0.y, w1.y};
    }
  }
  // inter-stage twiddle W_N^{sgn*n1*k2} for this lane's D-fragment elements
  float2 wT[TILES][8];
#pragma unroll
  for (int t = 0; t < TILES; ++t) {
    const int k2 = lr + 16 * t;
#pragma unroll
    for (int r = 0; r < 8; ++r) wT[t][r] = twf((r + hi8) * k2, N, sgn);
  }
  // Step C: A = DFT16: W16^{sgn*k1*n1}, M = k1 = lr, K = n1 (ai pre-negated)
  v2f arC[4], aiC[4], naiC[4];
#pragma unroll
  for (int c = 0; c < 4; ++c) {
    const int kk = 4 * c + hi2;
    float2 wa0 = twf(lr * kk, 16, sgn);
    float2 wa1 = twf(lr * (kk + 1), 16, sgn);
    arC[c] = v2f{wa0.x, wa1.x};
    aiC[c] = v2f{wa0.y, wa1.y};
    naiC[c] = v2f{-wa0.y, -wa1.y};
  }
  const v8f vzero = {0.f, 0.f, 0.f, 0.f, 0.f, 0.f, 0.f, 0.f};

  const int step = gridDim.x * 8;
  for (int gl = blockIdx.x * 8 + wave; gl < nlines; gl += step) {
    float2* base = buf + (long long)(gl / linesPerImage) * imageStride +
                   (long long)(gl % linesPerImage) * lineStride;

    // prefetch next line while this one computes (global_prefetch_b8)
    if (gl + step < nlines) {
      const int g2 = gl + step;
      const float2* nb = buf + (long long)(g2 / linesPerImage) * imageStride +
                         (long long)(g2 % linesPerImage) * lineStride;
      __builtin_prefetch(nb + (long long)(lane * (N / 32)) * elemStride, 0, 1);
    }

    for (int i = lane; i < N; i += 32) lds[i] = base[(long long)i * elemStride];
    __builtin_amdgcn_sched_barrier(0);

    // ---- Step A: Z[n1][k2] = sum_{n2<N2} x[n1+16*n2] * W_{N2}^{sgn*n2*k2}
    v8f zr[TILES], zi[TILES];
#pragma unroll
    for (int t = 0; t < TILES; ++t) { zr[t] = vzero; zi[t] = vzero; }

#pragma unroll
    for (int c = 0; c < CH_A; ++c) {
      const int kk = 4 * c + hi2;  // n2 for this half-wave
      float2 x0 = lds[lr + 16 * kk];
      float2 x1 = lds[lr + 16 * (kk + 1)];
      v2f axr = {x0.x, x1.x};
      v2f axi = {x0.y, x1.y};
      v2f naxi = {-x0.y, -x1.y};
#pragma unroll
      for (int t = 0; t < TILES; ++t) {
        zr[t] = wmma4(axr, brA[c][t], zr[t]);   // + xr*Wr
        zr[t] = wmma4(naxi, biA[c][t], zr[t]);  // - xi*Wi
        zi[t] = wmma4(axr, biA[c][t], zi[t]);   // + xr*Wi
        zi[t] = wmma4(axi, brA[c][t], zi[t]);   // + xi*Wr
      }
    }
    __builtin_amdgcn_sched_barrier(0);

    // ---- twiddle by W_N^{sgn*n1*k2}; store Z' at lds[n1 + 16*k2]
#pragma unroll
    for (int t = 0; t < TILES; ++t) {
      const int k2 = lr + 16 * t;
#pragma unroll
      for (int r = 0; r < 8; ++r) {
        const int n1 = r + hi8;
        float2 w = wT[t][r];
        float re = zr[t][r], im = zi[t][r];
        lds[n1 + 16 * k2] =
            make_float2(re * w.x - im * w.y, re * w.y + im * w.x);
      }
    }
    __builtin_amdgcn_sched_barrier(0);

    // ---- Step C: X[k2 + N2*k1] = sum_{n1<16} W16^{sgn*k1*n1} * Z'[n1][k2]
    v8f xr[TILES], xi[TILES];
#pragma unroll
    for (int t = 0; t < TILES; ++t) { xr[t] = vzero; xi[t] = vzero; }

#pragma unroll
    for (int c = 0; c < 4; ++c) {
      const int kk = 4 * c + hi2;  // n1
#pragma unroll
      for (int t = 0; t < TILES; ++t) {
        const int k2 = lr + 16 * t;
        float2 z0 = lds[kk + 16 * k2];
        float2 z1 = lds[kk + 1 + 16 * k2];
        v2f br = {z0.x, z1.x};
        v2f bi = {z0.y, z1.y};
        xr[t] = wmma4(arC[c], br, xr[t]);
        xr[t] = wmma4(naiC[c], bi, xr[t]);
        xi[t] = wmma4(arC[c], bi, xi[t]);
        xi[t] = wmma4(aiC[c], br, xi[t]);
      }
    }
    __builtin_amdgcn_sched_barrier(0);

#pragma unroll
    for (int t = 0; t < TILES; ++t) {
      const int k2 = lr + 16 * t;
#pragma unroll
      for (int r = 0; r < 8; ++r) {
        const int k1 = r + hi8;
        lds[k2 + N2 * k1] = make_float2(xr[t][r] * scale, xi[t][r] * scale);
      }
    }
    __builtin_amdgcn_sched_barrier(0);

    for (int i = lane; i < N; i += 32) base[(long long)i * elemStride] = lds[i];
    __builtin_amdgcn_sched_barrier(0);
  }
}

// ---------------------------------------------------------------------------
// Pointwise / setup kernels
// ---------------------------------------------------------------------------
__global__ void k_zero(float* p, long long n) {
  long long id = (long long)blockIdx.x * blockDim.x + threadIdx.x;
  if (id < n) p[id] = 0.f;
}

// mask3[i][j][l] for (512,512,3): mirrored-tiled einsum('fl,fij->ijl')
__global__ void k_build_mask(const float* mp, const float* cf, float* mask3) {
  long long id = (long long)blockIdx.x * blockDim.x + threadIdx.x;
  if (id >= 512LL * 512 * 3) return;
  int l = (int)(id % 3);
  long long t = id / 3;
  int j = (int)(t & 511), i = (int)(t >> 9);
  int ii = i & 7, jj = j & 7;
  int si = ii < 4 ? ii : 7 - ii;
  int sj = jj < 4 ? jj : 7 - jj;
  float acc = 0.f;
  for (int f = 0; f < 4; ++f) acc += cf[f * 3 + l] * mp[(f * 4 + si) * 4 + sj];
  mask3[id] = acc;
}

__global__ void k_height(const float* zc, const float* zv, float* hm) {
  long long id = (long long)blockIdx.x * blockDim.x + threadIdx.x;
  if (id >= 262144) return;
  float a = 0.f;
  for (int t = 0; t < 64; ++t) a += zc[t] * zv[(long long)t * 262144 + id];
  hm[id] = a;
}

// field = exp(i * kphase * hm) * aperture
__global__ void k_field(const float* hm, float2* u1, float kphase) {
  long long id = (long long)blockIdx.x * blockDim.x + threadIdx.x;
  if (id >= 262144) return;
  int i = (int)(id >> 9), j = (int)(id & 511);
  float dx = (float)(i - 256), dy = (float)(j - 256);
  if (dx * dx + dy * dy >= 65025.0f) {  // 255^2 aperture
    u1[id] = make_float2(0.f, 0.f);
    return;
  }
  float s, c;
  sincosf(kphase * hm[id], &s, &c);
  u1[id] = make_float2(c, s);
}

// angular-spectrum transfer, computed in fp64 (phase ~1e5 rad, like reference)
__global__ void k_transfer(float2* u, long long n, float wl, float dist) {
  long long id = (long long)blockIdx.x * blockDim.x + threadIdx.x;
  if (id >= n) return;
  int pix = (int)(id & 262143);
  int i = pix >> 9, j = pix & 511;
  const double fscale = 1.0 / (512.0 * 3.69e-6);
  double fi = (double)(i < 256 ? i : i - 512) * fscale;
  double fj = (double)(j < 256 ? j : j - 512) * fscale;
  double wfx = (double)wl * fi, wfy = (double)wl * fj;
  double arg = 1.0 - wfx * wfx - wfy * wfy;
  if (arg <= 0.0) {
    u[id] = make_float2(0.f, 0.f);
    return;
  }
  double kz = (6.283185307179586 / (double)wl) * sqrt(arg);
  double ph = (double)dist * kz;
  double s = sin(ph), c = cos(ph);
  float2 v = u[id];
  u[id] = make_float2((float)(v.x * c - v.y * s), (float)(v.x * s + v.y * c));
}

// U[k] = u1 * roll(mask_b, mi[k], mj[k])   (16 shifted copies)
__global__ void k_u2in(const float2* u1, const float* mask3, float2* U,
                       int band) {
  long long id = (long long)blockIdx.x * blockDim.x + threadIdx.x;
  if (id >= 16LL * 262144) return;
  int k = (int)(id >> 18);
  int pix = (int)(id & 262143);
  int i = pix >> 9, j = pix & 511;
  const int mov0 = (k >> 2) >= 2 ? -1 : 0;  // _mov = {0,0,-1,-1}
  const int mov1 = (k & 3) >= 2 ? -1 : 0;
  int si = (i - mov0 + 512) & 511;
  int sj = (j - mov1 + 512) & 511;
  float m = mask3[((long long)si * 512 + sj) * 3 + band];
  float2 v = u1[pix];
  U[id] = make_float2(v.x * m, v.y * m);
}

// 2x2-pooled |u3|^2 with deterministic per-block partial sums
__global__ __launch_bounds__(256) void k_psf(const float2* U, float* psf,
                                             float* partial) {
  long long id = (long long)blockIdx.x * 256 + threadIdx.x;  // 16*65536
  int k = (int)(id >> 16);
  int pq = (int)(id & 65535);
  int p = pq >> 8, q = pq & 255;
  const float2* ub = U + (long long)k * 262144;
  float acc = 0.f;
  for (int a = 0; a < 2; ++a)
    for (int b = 0; b < 2; ++b) {
      float2 v = ub[(long long)(2 * p + a) * 512 + (2 * q + b)];
      acc += v.x * v.x + v.y * v.y;
    }
  acc *= 0.25f;
  psf[id] = acc;
  __shared__ float red[256];
  red[threadIdx.x] = acc;
  __syncthreads();
  for (int s = 128; s > 0; s >>= 1) {
    if (threadIdx.x < s) red[threadIdx.x] += red[threadIdx.x + s];
    __syncthreads();
  }
  if (threadIdx.x == 0) partial[blockIdx.x] = red[0];  // 256 blocks per k
}

__global__ void k_sumreduce(const float* partial, float* sums) {
  int t = threadIdx.x;
  if (t >= 16) return;
  float a = 0.f;
  for (int i = 0; i < 256; ++i) a += partial[t * 256 + i];
  sums[t] = a;
}

// normalize + ifftshift (+128,+128) + pack to complex
__global__ void k_psf_pack(const float* psf, const float* sums, float2* otf) {
  long long id = (long long)blockIdx.x * blockDim.x + threadIdx.x;
  if (id >= 16LL * 65536) return;
  int k = (int)(id >> 16);
  int pq = (int)(id & 65535);
  int p = pq >> 8, q = pq & 255;
  int sp = (p + 128) & 255, sq = (q + 128) & 255;
  float v = psf[((long long)k << 16) + sp * 256 + sq] / sums[k];
  otf[id] = make_float2(v, 0.f);
}

// xs[n] = inputs[n,:,:,band] * smask_k  (comb i%4==ri, j%4==rj)
__global__ void k_xs(const float* inp, float2* xs, int band, int k) {
  long long id = (long long)blockIdx.x * blockDim.x + threadIdx.x;
  if (id >= 8LL * 65536) return;
  int n = (int)(id >> 16);
  int pq = (int)(id & 65535);
  int i = pq >> 8, j = pq & 255;
  int ri = k >> 2, rj = k & 3;
  float v = 0.f;
  if (((i & 3) == ri) && ((j & 3) == rj))
    v = inp[(((long long)n * 256 + i) * 256 + j) * 3 + band];
  xs[id] = make_float2(v, 0.f);
}

// acc += fft2(xs) * otf[k]
__global__ void k_macc(const float2* xs, const float2* otf, float2* acc,
                       int k) {
  long long id = (long long)blockIdx.x * blockDim.x + threadIdx.x;
  if (id >= 8LL * 65536) return;
  int pq = (int)(id & 65535);
  float2 a = xs[id];
  float2 b = otf[((long long)k << 16) + pq];
  float2 c = acc[id];
  acc[id] = make_float2(c.x + a.x * b.x - a.y * b.y,
                        c.y + a.x * b.y + a.y * b.x);
}

// out[n,i,j,c] += cf[fidx_c, band] * real(acc)   fidx = (0,1,3)
__global__ void k_out(const float2* acc, const float* cf, float* out,
                      int band) {
  long long id = (long long)blockIdx.x * blockDim.x + threadIdx.x;
  if (id >= 8LL * 65536) return;
  float v = acc[id].x;
  float w0 = cf[0 * 3 + band], w1 = cf[1 * 3 + band], w2 = cf[3 * 3 + band];
  long long o = id * 3;
  out[o + 0] += w0 * v;
  out[o + 1] += w1 * v;
  out[o + 2] += w2 * v;
}

// ---------------------------------------------------------------------------
// Host side
// ---------------------------------------------------------------------------
static inline int fft_blocks(int nlines) {
  int l8 = nlines / 8;
  return l8 > 256 ? 256 : l8;  // all our line counts are multiples of 2048 or 512
}
static inline void fft2_512(float2* buf, int batch, float sgn, float scale,
                            hipStream_t s) {
  int nlines = batch * 512;
  int blocks = fft_blocks(nlines);
  k_fft_pass<32><<<blocks, 256, 0, s>>>(buf, 512, 512LL * 512, 512, 1, sgn,
                                        scale, nlines);
  k_fft_pass<32><<<blocks, 256, 0, s>>>(buf, 512, 512LL * 512, 1, 512, sgn,
                                        scale, nlines);
}
static inline void fft2_256(float2* buf, int batch, float sgn, float scale,
                            hipStream_t s) {
  int nlines = batch * 256;
  int blocks = fft_blocks(nlines);
  k_fft_pass<16><<<blocks, 256, 0, s>>>(buf, 256, 256LL * 256, 256, 1, sgn,
                                        scale, nlines);
  k_fft_pass<16><<<blocks, 256, 0, s>>>(buf, 256, 256LL * 256, 1, 256, sgn,
                                        scale, nlines);
}

extern "C" void kernel_launch(void* const* d_in, const int* in_sizes, int n_in,
                              void* d_out, int out_size, void* d_ws,
                              size_t ws_size, hipStream_t stream) {
  const float* inp = (const float*)d_in[0];  // (8,256,256,3)
  const float* zc = (const float*)d_in[1];   // (64)
  const float* zv = (const float*)d_in[2];   // (64,512,512)
  const float* mp = (const float*)d_in[3];   // (4,4,4)
  const float* cf = (const float*)d_in[4];   // (4,3)
  float* out = (float*)d_out;                // (8,256,256,3)

  char* ws = (char*)d_ws;
  size_t off = 0;
  auto walloc = [&](size_t bytes) -> void* {
    void* p = ws + off;
    off = (off + bytes + 255) & ~(size_t)255;
    return p;
  };
  float2* U = (float2*)walloc(16LL * 262144 * sizeof(float2));    // 33.5 MB
  float2* u1 = (float2*)walloc(262144LL * sizeof(float2));        //  2.1 MB
  float* mask3 = (float*)walloc(512LL * 512 * 3 * sizeof(float)); //  3.1 MB
  float* hm = (float*)walloc(262144LL * sizeof(float));           //  1.0 MB
  float* psf = (float*)walloc(16LL * 65536 * sizeof(float));      //  4.2 MB
  float2* otf = (float2*)walloc(16LL * 65536 * sizeof(float2));   //  8.4 MB
  float2* xs = (float2*)walloc(8LL * 65536 * sizeof(float2));     //  4.2 MB
  float2* acc = (float2*)walloc(8LL * 65536 * sizeof(float2));    //  4.2 MB
  float* partial = (float*)walloc(4096 * sizeof(float));
  float* sums = (float*)walloc(16 * sizeof(float));

  const float wls[3] = {4.6e-7f, 5.4e-7f, 6.2e-7f};
  const float dns[3] = {0.52f, 0.515f, 0.51f};
  const float DIST_CODE = 0.047f;
  const float DIST_NET = 0.05f - 0.047f;  // fused -DIST_CODE then +DIST

  // zero output (accumulated across bands)
  k_zero<<<(8 * 65536 * 3 + 255) / 256, 256, 0, stream>>>(out, 8LL * 65536 * 3);
  k_build_mask<<<(512 * 512 * 3 + 255) / 256, 256, 0, stream>>>(mp, cf, mask3);
  k_height<<<1024, 256, 0, stream>>>(zc, zv, hm);

  for (int b = 0; b < 3; ++b) {
    const float wl = wls[b];

    // field -> u1 = prop(field, H(wl, DIST_CODE))
    k_field<<<1024, 256, 0, stream>>>(hm, u1, PI2F / wl * dns[b]);
    fft2_512(u1, 1, -1.f, 1.f, stream);
    k_transfer<<<1024, 256, 0, stream>>>(u1, 262144, wl, DIST_CODE);
    fft2_512(u1, 1, 1.f, 1.f / 512.f, stream);

    // u3 = ifft2( fft2(u1 * masks_s) * H(wl, DIST-DIST_CODE) )  (16 copies)
    k_u2in<<<16384, 256, 0, stream>>>(u1, mask3, U, b);
    fft2_512(U, 16, -1.f, 1.f, stream);
    k_transfer<<<16384, 256, 0, stream>>>(U, 16LL * 262144, wl, DIST_NET);
    fft2_512(U, 16, 1.f, 1.f / 512.f, stream);

    // psf -> normalized -> otf = fft2(ifftshift(psf))
    k_psf<<<4096, 256, 0, stream>>>(U, psf, partial);
    k_sumreduce<<<1, 16, 0, stream>>>(partial, sums);
    k_psf_pack<<<4096, 256, 0, stream>>>(psf, sums, otf);
    fft2_256(otf, 16, -1.f, 1.f, stream);

    // accumulate spec = sum_k fft2(x_b * smask_k) * otf_k
    k_zero<<<4096, 256, 0, stream>>>((float*)acc, 8LL * 65536 * 2);
    for (int k = 0; k < 16; ++k) {
      k_xs<<<2048, 256, 0, stream>>>(inp, xs, b, k);
      fft2_256(xs, 8, -1.f, 1.f, stream);
      k_macc<<<2048, 256, 0, stream>>>(xs, otf, acc, k);
    }
    fft2_256(acc, 8, 1.f, 1.f / 256.f, stream);
    k_out<<<2048, 256, 0, stream>>>(acc, cf, out, b);
  }
}